// RoiPooling_67070209295031
// MI455X (gfx1250) — compile-verified
//
#include <hip/hip_runtime.h>

// ROI bilinear pooling tuned for MI455X (gfx1250), memory-roofline driven:
//  - 411 MB write-once output -> non-temporal global_store_b128 (protect L2)
//  - 67 MB image is L2-resident -> gathers are L2/WGP$ hits
//  - scalar (wave-uniform) column-reuse cache halves read traffic:
//    adjacent pooled cells share corner columns; conditions depend only on
//    the ROI record (s_load) + px, so branches are s_cbranch, no divergence
//  - guarded global_prefetch_b8 warms the next *new* column one cell ahead

typedef float v4f __attribute__((ext_vector_type(4)));

#define POOL     14
#define IMG_W    128
#define IMG_C    1024
#define NUM_ROIS 512
#define ROWV     (IMG_C / 4)   // float4s per pixel

__global__ __launch_bounds__(256) void roi_bilinear_kernel(
    const float* __restrict__ img,
    const int*  __restrict__ rois,
    float* __restrict__ out)
{
    const int py  = blockIdx.x;   // pooled row  (0..13)
    const int roi = blockIdx.y;   // ROI index   (0..511)

    const int rx = rois[roi * 4 + 0];
    const int ry = rois[roi * 4 + 1];
    const int rw = rois[roi * 4 + 2];
    const int rh = rois[roi * 4 + 3];

    // Vertical coords, bit-exact vs reference: scale = h/P (f32 divide),
    // f = py*scale, i0 = floor(f), fy = f - i0, i1 = min(i0+1, h-1)
    const float yscale = (float)rh / (float)POOL;
    const float fyv    = (float)py * yscale;
    const int   iy0    = (int)floorf(fyv);
    const float fy     = fyv - (float)iy0;
    const int   iy1    = min(iy0 + 1, rh - 1);

    // 256 threads x float4 = 1024 channels, fully coalesced along C.
    const int c4 = threadIdx.x;
    const v4f* __restrict__ r0 =
        (const v4f*)(img + (size_t)(ry + iy0) * IMG_W * IMG_C) + c4;
    const v4f* __restrict__ r1 =
        (const v4f*)(img + (size_t)(ry + iy1) * IMG_W * IMG_C) + c4;
    v4f* __restrict__ o =
        (v4f*)(out + ((size_t)roi * POOL + py) * POOL * IMG_C) + c4;

    const float xscale = (float)rw / (float)POOL;

    // Wave-uniform column cache: raw top/bottom vectors for the last two
    // source columns (xa -> ta/ba, xb -> tb/bb).
    int xa = -1, xb = -1;
    v4f ta = {}, ba = {}, tb = {}, bb = {};

    #pragma unroll 1
    for (int px = 0; px < POOL; ++px) {
        const float fxv = (float)px * xscale;
        const int   ix0 = (int)floorf(fxv);
        const float fx  = fxv - (float)ix0;
        const int   ix1 = min(ix0 + 1, rw - 1);
        const int   xs0 = rx + ix0;
        const int   xs1 = rx + ix1;

        if (xs0 != xa) {                       // scalar branch
            if (xs0 == xb) { ta = tb; ba = bb; }
            else {
                ta = r0[(size_t)xs0 * ROWV];
                ba = r1[(size_t)xs0 * ROWV];
            }
            xa = xs0;
        }
        if (xs1 != xb) {                       // scalar branch
            if (xs1 == xa) { tb = ta; bb = ba; }
            else {
                tb = r0[(size_t)xs1 * ROWV];
                bb = r1[(size_t)xs1 * ROWV];
            }
            xb = xs1;
        }

        // Prefetch only the genuinely-new column of the NEXT cell
        // (gfx1250 global_prefetch_b8), guarded by scalar reuse check.
        if (px + 1 < POOL) {
            const int nix0 = (int)floorf((float)(px + 1) * xscale);
            const int nix1 = min(nix0 + 1, rw - 1);
            const int nxs1 = rx + nix1;
            if (nxs1 != xa && nxs1 != xb) {
                __builtin_prefetch((const void*)&r0[(size_t)nxs1 * ROWV], 0, 3);
                __builtin_prefetch((const void*)&r1[(size_t)nxs1 * ROWV], 0, 3);
            }
        }

        // Reference lerp order: x first (top/bot), then y.
        const v4f top = ta + (tb - ta) * fx;
        const v4f bot = ba + (bb - ba) * fx;
        const v4f res = top + (bot - top) * fy;

        // Write-once output: non-temporal B128 store.
        __builtin_nontemporal_store(res, &o[(size_t)px * ROWV]);
    }
}

extern "C" void kernel_launch(void* const* d_in, const int* in_sizes, int n_in,
                              void* d_out, int out_size, void* d_ws, size_t ws_size,
                              hipStream_t stream) {
    (void)in_sizes; (void)n_in; (void)out_size; (void)d_ws; (void)ws_size;
    const float* img  = (const float*)d_in[0];   // [1,128,128,1024] f32
    const int*   rois = (const int*)d_in[1];     // [1,512,4] i32 (x,y,w,h)
    // d_in[2] = pool_size scalar (14) — fixed by the reference, hardcoded.
    float* out = (float*)d_out;                  // [512,14,14,1024] f32

    dim3 grid(POOL, NUM_ROIS);   // x = pooled row, y = ROI (no int divide)
    dim3 block(256);             // 8 wave32s; thread = 4 channels (float4)
    roi_bilinear_kernel<<<grid, block, 0, stream>>>(img, rois, out);
}